// Net_31903017074850
// MI455X (gfx1250) — compile-verified
//
#include <hip/hip_runtime.h>
#include <math.h>

typedef float v2f __attribute__((ext_vector_type(2)));
typedef float v8f __attribute__((ext_vector_type(8)));

#define AA   300   // annotation size
#define DD   64    // gnn hidden
#define LL   8     // propagation steps
#define GATE 192   // 3*DD

__device__ __forceinline__ float sigmoidf_(float x) { return 1.0f / (1.0f + expf(-x)); }

// A-fragment (16x4 f32) from LDS row-major tile with leading dim ld.
// lane l holds A[l%16][k0 + 2*(l/16) + j] in vgpr j.
__device__ __forceinline__ v2f ldsA(const float* s, int ld, int lane, int k0) {
  int r = lane & 15;
  int k = k0 + ((lane >> 4) << 1);
  v2f a; a.x = s[r * ld + k]; a.y = s[r * ld + k + 1]; return a;
}
// B-fragment (4x16 f32) from global row-major B[k][n], leading dim ldb.
__device__ __forceinline__ v2f glbB(const float* B, int ldb, int lane, int k0, int n0) {
  int n = n0 + (lane & 15);
  int k = k0 + ((lane >> 4) << 1);
  v2f b; b.x = B[(size_t)k * ldb + n]; b.y = B[(size_t)(k + 1) * ldb + n]; return b;
}
// Branchless column-guarded variant: clamp address, zero via mask multiply.
// Avoids EXEC save/restore branches in the inner WMMA loop.
__device__ __forceinline__ v2f glbB_clamp(const float* B, int ldb, int ncols, int lane, int k0, int n0) {
  int n = n0 + (lane & 15);
  float msk = (n < ncols) ? 1.0f : 0.0f;
  int nc = (n < ncols) ? n : (ncols - 1);
  int k = k0 + ((lane >> 4) << 1);
  v2f b;
  b.x = B[(size_t)k * ldb + nc] * msk;
  b.y = B[(size_t)(k + 1) * ldb + nc] * msk;
  return b;
}
#define WMMA_F32(acc, a, b) \
  acc = __builtin_amdgcn_wmma_f32_16x16x4_f32(false, a, false, b, (short)0, acc, false, false)

// --------------------------------------------------------------------------
// One-time: transpose GRU weights [192,64] -> [64,192] for coalesced B reads.
__global__ void k_transpose(const float* __restrict__ wi, const float* __restrict__ wh,
                            float* __restrict__ wiT, float* __restrict__ whT) {
  int i = blockIdx.x * 256 + threadIdx.x;
  const int n = GATE * DD;
  if (i < n) {
    int r = i / DD, c = i % DD;
    wiT[c * GATE + r] = wi[i];
  } else if (i < 2 * n) {
    int j = i - n;
    int r = j / DD, c = j % DD;
    whT[c * GATE + r] = wh[j];
  }
}

// One-time: exclusive prefix sums over G graphs (G=500 -> trivial serial).
__global__ void k_prefix(const int* __restrict__ offset, const int* __restrict__ cnt,
                         int* __restrict__ offs, int* __restrict__ cofs, int G) {
  if (threadIdx.x == 0 && blockIdx.x == 0) {
    int so = 0, sc = 0;
    for (int g = 0; g < G; ++g) {
      offs[g] = so; cofs[g] = sc;
      so += offset[g]; sc += cnt[g];
    }
    cofs[G] = sc;
  }
}

// --------------------------------------------------------------------------
// h = relu(relu(x @ W0 + b0) @ W1 + b1)   [N,300] -> [N,64], 16 rows/block.
__global__ __launch_bounds__(256) void k_reduce(const float* __restrict__ x,
    const float* __restrict__ W0, const float* __restrict__ b0,
    const float* __restrict__ W1, const float* __restrict__ b1,
    float* __restrict__ h, int N) {
  __shared__ float sx[16 * AA];
  __shared__ float st[16 * AA];
  int tid = threadIdx.x, lane = tid & 31, wave = tid >> 5;
  long row0 = (long)blockIdx.x * 16;
  for (int i = tid; i < 16 * AA; i += 256) {
    int r = i / AA, c = i % AA;
    long rr = row0 + r; if (rr >= N) rr = N - 1;
    sx[i] = x[rr * AA + c];
  }
  __syncthreads();
  // phase 1: 19 column tiles over 300 cols, K = 300.
  // Tiles with n0+16 <= 300 take the unguarded fast path; only the last
  // partial tile uses the branchless clamp+mask path.
  for (int t = wave; t < 19; t += 8) {
    int n0 = t * 16;
    v8f acc = {};
    if (n0 + 16 <= AA) {
      for (int k0 = 0; k0 < AA; k0 += 4) {
        v2f a = ldsA(sx, AA, lane, k0);
        v2f b = glbB(W0, AA, lane, k0, n0);
        WMMA_F32(acc, a, b);
      }
    } else {
      for (int k0 = 0; k0 < AA; k0 += 4) {
        v2f a = ldsA(sx, AA, lane, k0);
        v2f b = glbB_clamp(W0, AA, AA, lane, k0, n0);
        WMMA_F32(acc, a, b);
      }
    }
    int col = n0 + (lane & 15);
    if (col < AA) {
      float bias = b0[col];
#pragma unroll
      for (int j = 0; j < 8; ++j) {
        int r = j + ((lane >> 4) << 3);
        float v = acc[j] + bias;
        st[r * AA + col] = v > 0.f ? v : 0.f;
      }
    }
  }
  __syncthreads();
  // phase 2: 4 column tiles for [300,64]
  for (int t = wave; t < 4; t += 8) {
    int n0 = t * 16;
    v8f acc = {};
    for (int k0 = 0; k0 < AA; k0 += 4) {
      v2f a = ldsA(st, AA, lane, k0);
      v2f b = glbB(W1, DD, lane, k0, n0);
      WMMA_F32(acc, a, b);
    }
    int col = n0 + (lane & 15);
    float bias = b1[col];
#pragma unroll
    for (int j = 0; j < 8; ++j) {
      int r = j + ((lane >> 4) << 3);
      long rr = row0 + r;
      if (rr < N) {
        float v = acc[j] + bias;
        h[rr * DD + col] = v > 0.f ? v : 0.f;
      }
    }
  }
}

// --------------------------------------------------------------------------
// m = h @ w  ([N,64]@[64,64]); also zero agg rows. 16 rows/block, 4 waves.
__global__ __launch_bounds__(128) void k_m(const float* __restrict__ h,
    const float* __restrict__ w, float* __restrict__ m, float* __restrict__ agg, int N) {
  __shared__ float sh[16 * 65];
  int tid = threadIdx.x, lane = tid & 31, wave = tid >> 5;
  long row0 = (long)blockIdx.x * 16;
  for (int i = tid; i < 16 * DD; i += 128) {
    int r = i >> 6, c = i & 63;
    long rr = row0 + r;
    long rc = rr < N ? rr : N - 1;
    sh[r * 65 + c] = h[rc * DD + c];
    if (rr < N) agg[rr * DD + c] = 0.0f;
  }
  __syncthreads();
  int n0 = wave * 16;
  v8f acc = {};
  for (int k0 = 0; k0 < DD; k0 += 4) {
    v2f a = ldsA(sh, 65, lane, k0);
    v2f b = glbB(w, DD, lane, k0, n0);
    WMMA_F32(acc, a, b);
  }
  int col = n0 + (lane & 15);
#pragma unroll
  for (int j = 0; j < 8; ++j) {
    int r = j + ((lane >> 4) << 3);
    long rr = row0 + r;
    if (rr < N) m[rr * DD + col] = acc[j];
  }
}

// --------------------------------------------------------------------------
// agg[dst] += m[src]: 64 lanes per edge, 4 edges per 256-thread block.
__global__ __launch_bounds__(256) void k_scatter(const int* __restrict__ src,
    const int* __restrict__ dst, const float* __restrict__ m,
    float* __restrict__ agg, int E) {
  int e = blockIdx.x * 4 + (threadIdx.x >> 6);
  int d = threadIdx.x & 63;
  if (e < E) {
    long s = src[e], t = dst[e];
    float v = m[s * DD + d];
    unsafeAtomicAdd(&agg[t * DD + d], v);
  }
}

// --------------------------------------------------------------------------
// GRU cell, in-place h update. 16 rows/block: 24 WMMA tiles (12 gi + 12 gh).
__global__ __launch_bounds__(256) void k_gru(float* __restrict__ h,
    const float* __restrict__ agg,
    const float* __restrict__ wiT, const float* __restrict__ whT,
    const float* __restrict__ bi, const float* __restrict__ bh, int N) {
  __shared__ float sa[16 * 65];
  __shared__ float sh[16 * 65];
  __shared__ float gi[16 * GATE];
  __shared__ float gh[16 * GATE];
  int tid = threadIdx.x, lane = tid & 31, wave = tid >> 5;
  long row0 = (long)blockIdx.x * 16;
  for (int i = tid; i < 16 * DD; i += 256) {
    int r = i >> 6, c = i & 63;
    long rr = row0 + r; if (rr >= N) rr = N - 1;
    sa[r * 65 + c] = agg[rr * DD + c];
    sh[r * 65 + c] = h[rr * DD + c];
  }
  __syncthreads();
  for (int T = wave; T < 24; T += 8) {
    const float* Am = (T < 12) ? sa : sh;
    const float* Bm = (T < 12) ? wiT : whT;
    int n0 = ((T < 12) ? T : (T - 12)) * 16;
    v8f acc = {};
    for (int k0 = 0; k0 < DD; k0 += 4) {
      v2f a = ldsA(Am, 65, lane, k0);
      v2f b = glbB(Bm, GATE, lane, k0, n0);
      WMMA_F32(acc, a, b);
    }
    float* OUT = (T < 12) ? gi : gh;
    int col = n0 + (lane & 15);
#pragma unroll
    for (int j = 0; j < 8; ++j) {
      int r = j + ((lane >> 4) << 3);
      OUT[r * GATE + col] = acc[j];
    }
  }
  __syncthreads();
  for (int i = tid; i < 16 * DD; i += 256) {
    int r = i >> 6, c = i & 63;
    long rr = row0 + r;
    if (rr >= N) continue;
    float ir = gi[r * GATE + c]        + bi[c];
    float iz = gi[r * GATE + 64 + c]   + bi[64 + c];
    float in_= gi[r * GATE + 128 + c]  + bi[128 + c];
    float hr = gh[r * GATE + c]        + bh[c];
    float hz = gh[r * GATE + 64 + c]   + bh[64 + c];
    float hn = gh[r * GATE + 128 + c]  + bh[128 + c];
    float rg = sigmoidf_(ir + hr);
    float zg = sigmoidf_(iz + hz);
    float ng = tanhf(in_ + rg * hn);
    h[rr * DD + c] = (1.0f - zg) * ng + zg * sh[r * 65 + c];
  }
}

// --------------------------------------------------------------------------
// Readout: gather sigmoid(h[idx]) -> relu(.@Wl0+bl0) -> sigmoid(relu(.@Wl1+bl1))
__global__ __launch_bounds__(256) void k_final(const float* __restrict__ h,
    const int* __restrict__ source_nodes, const int* __restrict__ offs,
    const int* __restrict__ cofs, int G,
    const float* __restrict__ Wl0, const float* __restrict__ bl0,
    const float* __restrict__ Wl1, const float* __restrict__ bl1,
    float* __restrict__ out, int S) {
  __shared__ float ss[16 * 65];
  __shared__ float t0[16 * 65];
  int tid = threadIdx.x, lane = tid & 31, wave = tid >> 5;
  int srow0 = blockIdx.x * 16;
  for (int i = tid; i < 16 * DD; i += 256) {
    int r = i >> 6, c = i & 63;
    int s = srow0 + r;
    float v = 0.0f;
    if (s < S) {
      int lo = 0, hi = G - 1;           // largest g with cofs[g] <= s
      while (lo < hi) {
        int mid = (lo + hi + 1) >> 1;
        if (cofs[mid] <= s) lo = mid; else hi = mid - 1;
      }
      long idx = (long)source_nodes[s] + (long)offs[lo];
      v = sigmoidf_(h[idx * DD + c]);
    }
    ss[r * 65 + c] = v;
  }
  __syncthreads();
  if (wave < 4) {
    int n0 = wave * 16;
    v8f acc = {};
    for (int k0 = 0; k0 < DD; k0 += 4) {
      v2f a = ldsA(ss, 65, lane, k0);
      v2f b = glbB(Wl0, DD, lane, k0, n0);
      WMMA_F32(acc, a, b);
    }
    int col = n0 + (lane & 15);
    float bias = bl0[col];
#pragma unroll
    for (int j = 0; j < 8; ++j) {
      int r = j + ((lane >> 4) << 3);
      float v = acc[j] + bias;
      t0[r * 65 + col] = v > 0.f ? v : 0.f;
    }
  }
  __syncthreads();
  if (tid < 16) {
    int s = srow0 + tid;
    if (s < S) {
      float sum = bl1[0];
      for (int k = 0; k < DD; ++k) sum += t0[tid * 65 + k] * Wl1[k];
      sum = sum > 0.f ? sum : 0.f;
      out[s] = sigmoidf_(sum);
    }
  }
}

// --------------------------------------------------------------------------
extern "C" void kernel_launch(void* const* d_in, const int* in_sizes, int n_in,
                              void* d_out, int out_size, void* d_ws, size_t ws_size,
                              hipStream_t stream) {
  const float* x    = (const float*)d_in[0];
  const int*   eidx = (const int*)d_in[1];
  const int*   snod = (const int*)d_in[2];
  const int*   offp = (const int*)d_in[3];
  const int*   scnt = (const int*)d_in[4];
  const float* W0   = (const float*)d_in[5];
  const float* b0   = (const float*)d_in[6];
  const float* W1   = (const float*)d_in[7];
  const float* b1   = (const float*)d_in[8];
  const float* ggc  = (const float*)d_in[9];
  const float* wih  = (const float*)d_in[10];
  const float* whh  = (const float*)d_in[11];
  const float* bih  = (const float*)d_in[12];
  const float* bhh  = (const float*)d_in[13];
  const float* Wl0  = (const float*)d_in[14];
  const float* bl0  = (const float*)d_in[15];
  const float* Wl1  = (const float*)d_in[16];
  const float* bl1  = (const float*)d_in[17];

  const int N = in_sizes[0] / AA;
  const int E = in_sizes[1] / 2;
  const int S = in_sizes[2];
  const int G = in_sizes[3];

  float* ws  = (float*)d_ws;
  float* h   = ws;
  float* m   = h   + (size_t)N * DD;
  float* agg = m   + (size_t)N * DD;
  float* wiT = agg + (size_t)N * DD;
  float* whT = wiT + (size_t)GATE * DD;
  int* offs  = (int*)(whT + (size_t)GATE * DD);
  int* cofs  = offs + G;

  int nblk = (N + 15) / 16;

  k_transpose<<<(2 * GATE * DD + 255) / 256, 256, 0, stream>>>(wih, whh, wiT, whT);
  k_prefix<<<1, 1, 0, stream>>>(offp, scnt, offs, cofs, G);
  k_reduce<<<nblk, 256, 0, stream>>>(x, W0, b0, W1, b1, h, N);
  for (int l = 0; l < LL; ++l) {
    k_m<<<nblk, 128, 0, stream>>>(h, ggc + (size_t)l * DD * DD, m, agg, N);
    k_scatter<<<(E + 3) / 4, 256, 0, stream>>>(eidx, eidx + E, m, agg, E);
    k_gru<<<nblk, 256, 0, stream>>>(h, agg, wiT, whT, bih, bhh, N);
  }
  k_final<<<(S + 15) / 16, 256, 0, stream>>>(h, snod, offs, cofs, G,
                                             Wl0, bl0, Wl1, bl1, (float*)d_out, S);
}